// QuantizeOld_90787018703331
// MI455X (gfx1250) — compile-verified
//
#include <hip/hip_runtime.h>
#include <hip/hip_bf16.h>
#include <float.h>

typedef __attribute__((ext_vector_type(2))) float v2f;
typedef __attribute__((ext_vector_type(8))) float v8f;

#define N_VEC      16384     // 16*32*32 query vectors
#define E_DIM      256
#define NE         8192      // codebook entries
#define NT_TILES   (NE / 16) // 512 codebook n-tiles
#define OUT_ELEMS  4194304   // 16*256*32*32
#define CH_STRIDE  1024      // 32*32 plane stride between channels in z
#define B_STRIDE   262144    // 256*1024 per-batch stride in z
#define LDS_PITCH  260       // 256 + 4 pad floats -> conflict-free ds_load_b64, 16B aligned rows
#define TILE_BYTES 16384     // 16 rows * 256 floats * 4B in global memory
#define BUF_BYTES  (16 * LDS_PITCH * 4)

// ---------------- kernel 1: per-row squared norms of the codebook ----------------
__global__ __launch_bounds__(256) void vq_enorm_kernel(const float* __restrict__ emb,
                                                       float* __restrict__ enorm) {
    int r = blockIdx.x * 256 + threadIdx.x;
    if (r >= NE) return;
    const float4* row = (const float4*)(emb + (size_t)r * E_DIM);
    float s = 0.f;
#pragma unroll 8
    for (int i = 0; i < E_DIM / 4; ++i) {
        float4 v = row[i];
        s += v.x * v.x + v.y * v.y + v.z * v.z + v.w * v.w;
    }
    enorm[r] = s;
}

// ---------------- kernel 2: WMMA fp32 distance + argmin ----------------
// Block = 256 threads = 8 waves; wave w owns z-rows [ (blockIdx*8+w)*16 , +16 ).
// A-fragments (z tile, M=16 K=256) live in registers. Embedding n-tiles are
// DMA'd 16x256 into double-buffered padded LDS with GLOBAL_LOAD_ASYNC_TO_LDS_B128
// (ASYNCcnt-tracked) so the tile-(nt+1) copy overlaps the tile-nt WMMA chain.
// The K-reduction runs on 4 independent accumulator chains to break the
// WMMA D->C RAW dependency (fp32 WMMA has no TRANS co-execution on CDNA5).
__global__ __launch_bounds__(256) void vq_argmin_kernel(const float* __restrict__ z,
                                                        const float* __restrict__ emb,
                                                        const float* __restrict__ enorm,
                                                        int* __restrict__ idx_out) {
    __shared__ float etile[2][16 * LDS_PITCH];

    const int tid  = threadIdx.x;
    const int lane = tid & 31;
    const int wave = tid >> 5;
    const int l15  = lane & 15;
    const int koff = (lane >> 4) << 1;          // fp32 WMMA A/B: lane-half selects K {0,2}

    const int m0 = (blockIdx.x * 8 + wave) * 16;
    const int m  = m0 + l15;                    // this lane's z row (A: lane = M)
    const int b  = m >> 10;
    const int p  = m & 1023;
    const float* zrow = z + (size_t)b * B_STRIDE + p;

    // Per-thread async-copy addressing: 4 float4 chunks of the 16x256 tile.
    unsigned long long gbase[4];                // global byte address for tile 0
    unsigned           lbase[4];                // LDS byte address in buffer 0
#pragma unroll
    for (int j = 0; j < 4; ++j) {
        int g    = j * 256 + tid;               // float4 index in tile: 0..1023
        int row  = g >> 6;                      // 64 float4 per row
        int col4 = g & 63;
        gbase[j] = (unsigned long long)(const void*)emb + (unsigned long long)g * 16ull;
        lbase[j] = (unsigned)(unsigned long long)(const void*)&etile[0][row * LDS_PITCH + col4 * 4];
    }

    // Load A fragments for the whole K=256: a[c] covers K = 4c+koff, 4c+koff+1.
    v2f a[64];
#pragma unroll
    for (int c = 0; c < 64; ++c) {
        int k0 = c * 4 + koff;
        v2f t;
        t.x = zrow[(size_t)k0 * CH_STRIDE];
        t.y = zrow[(size_t)(k0 + 1) * CH_STRIDE];
        a[c] = t;
    }

    float bestd[8];
    int   besti[8];
#pragma unroll
    for (int i = 0; i < 8; ++i) { bestd[i] = FLT_MAX; besti[i] = 0; }

    // Kick off the DMA of tile 0 into buffer 0.
#pragma unroll
    for (int j = 0; j < 4; ++j) {
        asm volatile("global_load_async_to_lds_b128 %0, %1, off"
                     :: "v"(lbase[j]), "v"(gbase[j]) : "memory");
    }

    for (int nt = 0; nt < NT_TILES; ++nt) {
        const int cur = nt & 1;
        // Tile nt has landed (own loads done), then barrier covers all waves' chunks.
        asm volatile("s_wait_asynccnt 0x0" ::: "memory");
        __syncthreads();
        // Start the DMA of tile nt+1 into the other buffer (safe: the barrier above
        // guarantees every wave finished computing from that buffer at iter nt-1).
        if (nt + 1 < NT_TILES) {
#pragma unroll
            for (int j = 0; j < 4; ++j) {
                unsigned long long ga = gbase[j] + (unsigned long long)(nt + 1) * TILE_BYTES;
                unsigned           la = lbase[j] + (unsigned)((cur ^ 1) * BUF_BYTES);
                asm volatile("global_load_async_to_lds_b128 %0, %1, off"
                             :: "v"(la), "v"(ga) : "memory");
            }
        }

        // 4 independent accumulator chains over the K dimension.
        v8f acc0 = {0.f, 0.f, 0.f, 0.f, 0.f, 0.f, 0.f, 0.f};
        v8f acc1 = {0.f, 0.f, 0.f, 0.f, 0.f, 0.f, 0.f, 0.f};
        v8f acc2 = {0.f, 0.f, 0.f, 0.f, 0.f, 0.f, 0.f, 0.f};
        v8f acc3 = {0.f, 0.f, 0.f, 0.f, 0.f, 0.f, 0.f, 0.f};
        const float* brow = &etile[cur][l15 * LDS_PITCH + koff];  // B: lane = N, half selects K
#pragma unroll
        for (int c = 0; c < 64; c += 4) {
            v2f b0 = *(const v2f*)(brow + (c + 0) * 4);
            v2f b1 = *(const v2f*)(brow + (c + 1) * 4);
            v2f b2 = *(const v2f*)(brow + (c + 2) * 4);
            v2f b3 = *(const v2f*)(brow + (c + 3) * 4);
            acc0 = __builtin_amdgcn_wmma_f32_16x16x4_f32(
                false, a[c + 0], false, b0, (short)0, acc0, false, false);
            acc1 = __builtin_amdgcn_wmma_f32_16x16x4_f32(
                false, a[c + 1], false, b1, (short)0, acc1, false, false);
            acc2 = __builtin_amdgcn_wmma_f32_16x16x4_f32(
                false, a[c + 2], false, b2, (short)0, acc2, false, false);
            acc3 = __builtin_amdgcn_wmma_f32_16x16x4_f32(
                false, a[c + 3], false, b3, (short)0, acc3, false, false);
        }

        const int   nIdx = nt * 16 + l15;
        const float en   = enorm[nIdx];
#pragma unroll
        for (int i = 0; i < 8; ++i) {
            float s = (acc0[i] + acc1[i]) + (acc2[i] + acc3[i]);
            float d = en - 2.0f * s;            // |z|^2 constant per row: drop it
            if (d < bestd[i]) { bestd[i] = d; besti[i] = nIdx; }  // strict < keeps first index
        }
    }

    // Cross-lane argmin within each 16-lane group (C layout: lane = N, half picks M or M+8).
#pragma unroll
    for (int i = 0; i < 8; ++i) {
        float d  = bestd[i];
        int   ix = besti[i];
#pragma unroll
        for (int msk = 1; msk < 16; msk <<= 1) {
            float od = __shfl_xor(d, msk, 16);
            int   oi = __shfl_xor(ix, msk, 16);
            if (od < d || (od == d && oi < ix)) { d = od; ix = oi; }
        }
        if (l15 == 0) {
            int row = i + ((lane >> 4) << 3);   // half 0 -> rows 0..7, half 1 -> 8..15
            idx_out[m0 + row] = ix;
        }
    }
}

// ---------------- kernel 3: gather z_q, straight-through out, loss partials, indices ----------------
__global__ __launch_bounds__(256) void vq_gather_kernel(const float* __restrict__ z,
                                                        const float* __restrict__ emb,
                                                        const int* __restrict__ idx,
                                                        float* __restrict__ out,
                                                        float* __restrict__ idx_f,
                                                        float* __restrict__ partial) {
    __shared__ float red[256];
    const int t  = threadIdx.x;
    const int nl = t & 15;          // vector within the 16-vector group (coalesced z/out)
    const int cg = t >> 4;          // channel phase
    const int n0 = blockIdx.x * 16;
    const int n  = n0 + nl;
    const int id = idx[n];
    const int b  = n >> 10;
    const int p  = n & 1023;
    const size_t zbase = (size_t)b * B_STRIDE + p;
    const float* erow  = emb + (size_t)id * E_DIM;

    float acc = 0.f;
#pragma unroll
    for (int c0 = 0; c0 < E_DIM; c0 += 16) {
        int c = c0 + cg;
        float e  = erow[c];
        size_t za = zbase + (size_t)c * CH_STRIDE;
        float zv = z[za];
        out[za] = zv + (e - zv);    // straight-through: z + sg(z_q - z)
        float dd = e - zv;
        acc += dd * dd;
    }
    red[t] = acc;
    __syncthreads();
    for (int s = 128; s > 0; s >>= 1) {
        if (t < s) red[t] += red[t + s];
        __syncthreads();
    }
    if (t == 0) partial[blockIdx.x] = red[0];
    if (t < 16) idx_f[n0 + t] = (float)idx[n0 + t];
}

// ---------------- kernel 4: deterministic loss finalization ----------------
__global__ __launch_bounds__(256) void vq_loss_kernel(const float* __restrict__ partial,
                                                      float* __restrict__ loss) {
    __shared__ float red[256];
    const int t = threadIdx.x;
    float s = 0.f;
    for (int i = t; i < 1024; i += 256) s += partial[i];
    red[t] = s;
    __syncthreads();
    for (int st = 128; st > 0; st >>= 1) {
        if (t < st) red[t] += red[t + st];
        __syncthreads();
    }
    if (t == 0) *loss = red[0] * (1.25f / (float)OUT_ELEMS);  // (1 + BETA) * mean
}

extern "C" void kernel_launch(void* const* d_in, const int* in_sizes, int n_in,
                              void* d_out, int out_size, void* d_ws, size_t ws_size,
                              hipStream_t stream) {
    const float* z   = (const float*)d_in[0];   // (16,256,32,32) fp32
    const float* emb = (const float*)d_in[1];   // (8192,256)    fp32

    float* out   = (float*)d_out;               // [0, 4194304): z_q_st (B,C,H,W)
    float* loss  = out + OUT_ELEMS;             // [4194304]:    scalar loss
    float* idx_f = out + OUT_ELEMS + 1;         // [+1, +16385): indices as float

    char*  ws      = (char*)d_ws;
    float* enorm   = (float*)ws;                         //  8192 f32 = 32 KB
    int*   idx     = (int*)(ws + 32768);                 // 16384 i32 = 64 KB
    float* partial = (float*)(ws + 32768 + 65536);       //  1024 f32 =  4 KB

    vq_enorm_kernel <<<NE / 256,     256, 0, stream>>>(emb, enorm);
    vq_argmin_kernel<<<N_VEC / 128,  256, 0, stream>>>(z, emb, enorm, idx);
    vq_gather_kernel<<<N_VEC / 16,   256, 0, stream>>>(z, emb, idx, out, idx_f, partial);
    vq_loss_kernel  <<<1,            256, 0, stream>>>(partial, loss);
}